// STGCN_9002251452898
// MI455X (gfx1250) — compile-verified
//
#include <hip/hip_runtime.h>
#include <hip/hip_bf16.h>
#include <math.h>

typedef float v2f __attribute__((ext_vector_type(2)));
typedef float v8f __attribute__((ext_vector_type(8)));

#define BM 256
#define BN 64
#define BK 32
#define LDA_S 36   // dword stride per A row (32 data + 4 pad); even -> aligned b64 frags

// C[M x (2*Nsplit)] = act(A)[M x K] @ [W0 | W1], optional embedding add on A,
// optional bias on the W0 half. Row-major everywhere. M%BM==0, (2*Nsplit)%BN==0.
// Each of 8 waves computes a 32x64 sub-tile: 2 row-tiles x 4 col-tiles of 16x16.
__global__ __launch_bounds__(256)
void gemm_wmma_f32(const float* __restrict__ A, int M, int K,
                   const float* __restrict__ W0, const float* __restrict__ W1,
                   int Nsplit, const float* __restrict__ bias,
                   const float* __restrict__ Temb, const float* __restrict__ Semb,
                   float* __restrict__ Cout, int ldc)
{
    __shared__ __align__(16) float As[BM * LDA_S];          // 36864 B
    __shared__ __align__(16) float Bs[(BK / 4) * BN * 4];   // 8192 B, fragment-major

    const int tid     = threadIdx.x;
    const int rowBase = blockIdx.y * BM;
    const int colBase = blockIdx.x * BN;
    const bool isW0   = (colBase < Nsplit);
    const float* Wsel = isW0 ? W0 : W1;
    const int colOff  = isW0 ? colBase : (colBase - Nsplit);

    const int w     = tid >> 5;         // wave 0..7 -> 32-row strip
    const int lane  = tid & 31;
    const int lhalf = lane & 15;
    const int h     = lane >> 4;        // 0: K+{0,1}; 1: K+{2,3}

    v8f acc[2][4];
    #pragma unroll
    for (int m = 0; m < 2; ++m)
        #pragma unroll
        for (int ct = 0; ct < 4; ++ct)
            acc[m][ct] = (v8f){0,0,0,0,0,0,0,0};

    for (int kBase = 0; kBase < K; kBase += BK) {
        // ---- stage A tile (fused embedding add), coalesced over k ----
        for (int i = tid; i < BM * BK; i += 256) {
            int r = i >> 5, c = i & 31;
            int k = kBase + c;
            float v = 0.f;
            if (k < K) {
                int grow = rowBase + r;
                v = A[(long)grow * K + k];
                if (Temb) {
                    int nn = grow % 36;
                    v += Temb[(nn / 12) * K + k] + Semb[(nn % 12) * K + k];
                }
            }
            As[r * LDA_S + c] = v;
        }
        // ---- stage B tile, packed so each operand is one aligned b64 ----
        // dword index = g*256 + c*4 + 2*h + comp  (g = k-group of 4, h = pair)
        for (int i = tid; i < BK * BN; i += 256) {
            int kr = i >> 6, c = i & 63;
            int k = kBase + kr;
            float v = (k < K) ? Wsel[(long)k * Nsplit + colOff + c] : 0.f;
            int g = kr >> 2, hh = (kr >> 1) & 1, comp = kr & 1;
            Bs[g * 256 + c * 4 + hh * 2 + comp] = v;
        }
        __syncthreads();

        #pragma unroll
        for (int g = 0; g < 8; ++g) {
            const int aOff = (w * 32 + lhalf) * LDA_S + g * 4 + 2 * h;
            v2f a0 = *(const v2f*)&As[aOff];
            v2f a1 = *(const v2f*)&As[aOff + 16 * LDA_S];
            #pragma unroll
            for (int ct = 0; ct < 4; ++ct) {
                v2f b = *(const v2f*)&Bs[g * 256 + (ct * 16 + lhalf) * 4 + 2 * h];
                acc[0][ct] = __builtin_amdgcn_wmma_f32_16x16x4_f32(false, a0, false, b, (short)0, acc[0][ct], false, false);
                acc[1][ct] = __builtin_amdgcn_wmma_f32_16x16x4_f32(false, a1, false, b, (short)0, acc[1][ct], false, false);
            }
        }
        __syncthreads();
    }

    // ---- epilogue: C/D layout vgpr r -> (row r | r+8, col lane&15) ----
    #pragma unroll
    for (int m = 0; m < 2; ++m) {
        const int rowW = rowBase + w * 32 + m * 16 + ((lane < 16) ? 0 : 8);
        #pragma unroll
        for (int ct = 0; ct < 4; ++ct) {
            int col = colBase + ct * 16 + lhalf;
            float badd = (bias != nullptr && isW0) ? bias[col] : 0.f;
            v8f a = acc[m][ct];
            #pragma unroll
            for (int r = 0; r < 8; ++r)
                Cout[(long)(rowW + r) * ldc + col] = a[r] + badd;
        }
    }
}

__global__ void deg_kernel(const int* __restrict__ ei, int* __restrict__ deg, int E) {
    int e = blockIdx.x * blockDim.x + threadIdx.x;
    if (e < E) atomicAdd(&deg[ei[e]], 1);
}

__global__ void dinv_kernel(const int* __restrict__ deg, float* __restrict__ dinv, int n) {
    int i = blockIdx.x * blockDim.x + threadIdx.x;
    if (i < n) { int d = deg[i]; dinv[i] = (d > 0) ? rsqrtf((float)d) : 0.f; }
}

// C[dst, 0:Hf] += -dinv[src]*dinv[dst] * C[src, srcOff:srcOff+Hf]
__global__ void scatter_kernel(const int* __restrict__ ei, const float* __restrict__ dinv,
                               float* __restrict__ Cbuf, int ldc, int srcOff, int Hf, int E)
{
    int epb = blockDim.x / Hf;
    int e = blockIdx.x * epb + threadIdx.x / Hf;
    int f = threadIdx.x % Hf;
    if (e >= E) return;
    int src = ei[e];
    int dst = ei[E + e];
    float w = -dinv[src] * dinv[dst];
    float v = w * Cbuf[(long)src * ldc + srcOff + f];
    atomicAdd(&Cbuf[(long)dst * ldc + f], v);
}

__global__ void bn_stats(const float* __restrict__ C, int ldc, int Hc, int M,
                         float* __restrict__ sum, float* __restrict__ sumsq)
{
    int col = threadIdx.x % Hc;
    int rpb = blockDim.x / Hc;
    int row = blockIdx.x * rpb + threadIdx.x / Hc;
    int stride = gridDim.x * rpb;
    float s = 0.f, q = 0.f;
    for (; row < M; row += stride) {
        float v = C[(long)row * ldc + col];
        s += v; q += v * v;
    }
    atomicAdd(&sum[col], s);
    atomicAdd(&sumsq[col], q);
}

__global__ void bn_apply(const float* __restrict__ Cin, int ldc, int Hc, int M,
                         const float* __restrict__ sum, const float* __restrict__ sumsq,
                         const float* __restrict__ g, const float* __restrict__ bt,
                         float* __restrict__ Y)
{
    long i = (long)blockIdx.x * blockDim.x + threadIdx.x;
    if (i >= (long)M * Hc) return;
    int col = (int)(i % Hc);
    long row = i / Hc;
    float invM = 1.f / (float)M;
    float mu = sum[col] * invM;
    float var = sumsq[col] * invM - mu * mu;
    float x = Cin[row * ldc + col];
    float y = (x - mu) * rsqrtf(var + 1e-5f) * g[col] + bt[col];
    Y[i] = fmaxf(y, 0.f);
}

// feat[b, 0:64] = max over 12 nodes, feat[b, 64:128] = mean  (nodes b*36+12..23)
__global__ void pool_kernel(const float* __restrict__ Y2, float* __restrict__ feat) {
    int b = blockIdx.x, f = threadIdx.x; // 64 threads
    const float* base = Y2 + ((long)b * 36 + 12) * 64 + f;
    float mx = -3.4e38f, sm = 0.f;
    #pragma unroll
    for (int s = 0; s < 12; ++s) { float v = base[s * 64]; mx = fmaxf(mx, v); sm += v; }
    feat[b * 128 + f] = mx;
    feat[b * 128 + 64 + f] = sm * (1.f / 12.f);
}

// writes outF[0,5120), outD[5120,10240), fusion[10240,272384), out[272384,277504)
__global__ void head_kernel(const float* __restrict__ featF, const float* __restrict__ featD,
                            const float* __restrict__ WlinF, const float* __restrict__ blinF,
                            const float* __restrict__ WlinD, const float* __restrict__ blinD,
                            const float* __restrict__ Wlin, const float* __restrict__ blin,
                            float* __restrict__ out)
{
    int b = blockIdx.x, t = threadIdx.x; // 256 threads
    __shared__ float fus[256];
    __shared__ float logits[15];
    fus[t] = (t < 128) ? featF[b * 128 + t] : featD[b * 128 + (t - 128)];
    __syncthreads();
    out[10240 + b * 256 + t] = fus[t];
    if (t < 5) {
        float aF = blinF[t], aD = blinD[t], aA = blin[t];
        for (int i = 0; i < 128; ++i) {
            aF += fus[i] * WlinF[i * 5 + t];
            aD += fus[128 + i] * WlinD[i * 5 + t];
        }
        for (int i = 0; i < 256; ++i) aA += fus[i] * Wlin[i * 5 + t];
        logits[t] = aF; logits[5 + t] = aD; logits[10 + t] = aA;
    }
    __syncthreads();
    if (t < 3) {
        float L[5];
        float m = -3.4e38f;
        for (int i = 0; i < 5; ++i) { L[i] = logits[t * 5 + i]; m = fmaxf(m, L[i]); }
        float s = 0.f;
        for (int i = 0; i < 5; ++i) { L[i] = expf(L[i] - m); s += L[i]; }
        float inv = 1.f / s;
        float* o = out + ((t == 0) ? 0 : (t == 1) ? 5120 : 272384) + b * 5;
        for (int i = 0; i < 5; ++i) o[i] = L[i] * inv;
    }
}

static inline int cdiv(int a, int b) { return (a + b - 1) / b; }

extern "C" void kernel_launch(void* const* d_in, const int* in_sizes, int n_in,
                              void* d_out, int out_size, void* d_ws, size_t ws_size,
                              hipStream_t stream)
{
    const int N = 36864, K1 = 1025, E = 589824;

    // workspace layout (floats)
    float* ws    = (float*)d_ws;
    float* Cbuf  = ws;                          // N*512
    float* Y1    = Cbuf  + (size_t)N * 512;     // N*256
    float* G2    = Y1    + (size_t)N * 256;     // N*128
    float* Y2    = G2    + (size_t)N * 128;     // N*64
    float* featF = Y2    + (size_t)N * 64;      // 1024*128
    float* featD = featF + 131072;              // 1024*128
    float* dinv  = featD + 131072;              // N
    int*   deg   = (int*)(dinv + N);            // N
    float* bns   = (float*)(deg + N);           // 256
    float* bnq   = bns + 256;                   // 256

    const float* xs[2]    = {(const float*)d_in[0], (const float*)d_in[1]};
    const int*   eis[2]   = {(const int*)d_in[2],   (const int*)d_in[3]};
    const float* Tembs[2] = {(const float*)d_in[4], (const float*)d_in[6]};
    const float* Sembs[2] = {(const float*)d_in[5], (const float*)d_in[7]};

    for (int br = 0; br < 2; ++br) {
        int p = br ? 18 : 8;
        const float* W0a = (const float*)d_in[p + 0];
        const float* W1a = (const float*)d_in[p + 1];
        const float* ba  = (const float*)d_in[p + 2];
        const float* ga  = (const float*)d_in[p + 3];
        const float* bta = (const float*)d_in[p + 4];
        const float* W0b = (const float*)d_in[p + 5];
        const float* W1b = (const float*)d_in[p + 6];
        const float* bb  = (const float*)d_in[p + 7];
        const float* gb  = (const float*)d_in[p + 8];
        const float* btb = (const float*)d_in[p + 9];
        const float* x   = xs[br];
        const int*   ei  = eis[br];
        float* feat = br ? featD : featF;

        // degree -> dinv
        hipMemsetAsync(deg, 0, (size_t)N * sizeof(int), stream);
        deg_kernel<<<cdiv(E, 256), 256, 0, stream>>>(ei, deg, E);
        dinv_kernel<<<cdiv(N, 256), 256, 0, stream>>>(deg, dinv, N);

        // layer 1: (x+emb) @ [W0|W1] -> Cbuf (N x 512), bias on first 256 cols
        gemm_wmma_f32<<<dim3(512 / BN, N / BM), 256, 0, stream>>>(
            x, N, K1, W0a, W1a, 256, ba, Tembs[br], Sembs[br], Cbuf, 512);
        scatter_kernel<<<E, 256, 0, stream>>>(ei, dinv, Cbuf, 512, 256, 256, E);
        hipMemsetAsync(bns, 0, 512 * sizeof(float), stream);
        bn_stats<<<288, 256, 0, stream>>>(Cbuf, 512, 256, N, bns, bnq);
        bn_apply<<<(N * 256) / 256, 256, 0, stream>>>(Cbuf, 512, 256, N, bns, bnq, ga, bta, Y1);

        // layer 2: Y1 @ [W0|W1] -> G2 (N x 128), bias on first 64 cols
        gemm_wmma_f32<<<dim3(128 / BN, N / BM), 256, 0, stream>>>(
            Y1, N, 256, W0b, W1b, 64, bb, nullptr, nullptr, G2, 128);
        scatter_kernel<<<E / 4, 256, 0, stream>>>(ei, dinv, G2, 128, 64, 64, E);
        hipMemsetAsync(bns, 0, 512 * sizeof(float), stream);
        bn_stats<<<288, 256, 0, stream>>>(G2, 128, 64, N, bns, bnq);
        bn_apply<<<(N * 64) / 256, 256, 0, stream>>>(G2, 128, 64, N, bns, bnq, gb, btb, Y2);

        pool_kernel<<<1024, 64, 0, stream>>>(Y2, feat);
    }

    head_kernel<<<1024, 256, 0, stream>>>(
        featF, featD,
        (const float*)d_in[28], (const float*)d_in[29],
        (const float*)d_in[30], (const float*)d_in[31],
        (const float*)d_in[32], (const float*)d_in[33],
        (float*)d_out);
}